// KernelORF_33689723469901
// MI455X (gfx1250) — compile-verified
//
#include <hip/hip_runtime.h>

typedef float v2f __attribute__((ext_vector_type(2)));
typedef float v8f __attribute__((ext_vector_type(8)));

constexpr int L = 4096, N = 16, H = 16, E = 64, F = 64;
constexpr float NORMALIZER = 0.35355339059327373f; // 64^(-0.25) = 2^(-1.5)
constexpr float KEPS = 1e-6f;

// ---------------------------------------------------------------------------
// Kernel A: per-head modified Gram-Schmidt on the ROWS of projection_matrix[h].
// This equals QR(mat^T) with the sign fix from the reference (R diag > 0).
// One wave (32 lanes) per head; lane t owns elements t and t+32 of each row.
// ---------------------------------------------------------------------------
__global__ __launch_bounds__(32) void orth_mgs_kernel(const float* __restrict__ proj,
                                                      float* __restrict__ W) {
  __shared__ __align__(16) float M[64 * 64];
  const int h = blockIdx.x;
  const int t = threadIdx.x; // 0..31

  const float* src = proj + (size_t)h * (E * E);
  // Load 64x64 floats cooperatively as float4 (1024 float4, 32 lanes).
  for (int i = t; i < (E * E) / 4; i += 32) {
    ((float4*)M)[i] = ((const float4*)src)[i];
  }
  __syncthreads();

  for (int j = 0; j < 64; ++j) {
    float v0 = M[j * 64 + t];
    float v1 = M[j * 64 + t + 32];
    for (int i = 0; i < j; ++i) {
      const float w0 = M[i * 64 + t];
      const float w1 = M[i * 64 + t + 32];
      float d = w0 * v0 + w1 * v1;
      d += __shfl_xor(d, 1);
      d += __shfl_xor(d, 2);
      d += __shfl_xor(d, 4);
      d += __shfl_xor(d, 8);
      d += __shfl_xor(d, 16);
      v0 -= d * w0;
      v1 -= d * w1;
    }
    float nn = v0 * v0 + v1 * v1;
    nn += __shfl_xor(nn, 1);
    nn += __shfl_xor(nn, 2);
    nn += __shfl_xor(nn, 4);
    nn += __shfl_xor(nn, 8);
    nn += __shfl_xor(nn, 16);
    const float rinv = rsqrtf(fmaxf(nn, 1e-30f));
    v0 *= rinv;
    v1 *= rinv;
    M[j * 64 + t] = v0;
    M[j * 64 + t + 32] = v1;
    __syncthreads();
    W[(size_t)h * (F * E) + j * 64 + t] = v0;
    W[(size_t)h * (F * E) + j * 64 + t + 32] = v1;
  }
}

// ---------------------------------------------------------------------------
// Kernel B: fused  out[l,n,h,f] = exp( <x_l, W_f> - 0.5*||x_l||^2 ) + eps
// with x = data * normalizer. Batched (n,h) GEMM via V_WMMA_F32_16X16X4_F32.
// Block = 256 thr (8 waves). Wave -> 16 rows of l, all 64 f (4 WMMA tiles).
// W[h] staged in LDS, padded row stride 66 floats (8B-aligned float2 reads).
// data/out are streamed exactly once and exceed the 192MB L2 combined, so
// the global accesses are non-temporal; only W[h] (in LDS) is reused.
//
// f32 WMMA fragment layout (lane = 32 lanes, m = lane&15, half = lane>>4):
//   A (16x4):  a[j] = A[m][2*half + j]       -> consecutive float2 in memory
//   B (4x16):  b[j] = B[2*half + j][m]       -> = W[f0+m][k0+2*half + j]
//   C/D:       d[j] -> row (j + 8*half), col m
// ---------------------------------------------------------------------------
__global__ __launch_bounds__(256) void favor_wmma_kernel(const float* __restrict__ data,
                                                         const float* __restrict__ W,
                                                         float* __restrict__ out) {
  __shared__ __align__(16) float Wl[F * 66];

  const int nh = blockIdx.y;
  const int n = nh / H;
  const int hd = nh % H;
  const int tid = threadIdx.x;

  // Stage W[hd] (f-major rows of length 64) into padded LDS.
  const float* Wsrc = W + (size_t)hd * (F * E);
  for (int i = tid; i < (F * E) / 4; i += 256) {
    const float4 v = ((const float4*)Wsrc)[i];
    const int flat = i * 4;
    const int f = flat >> 6;
    const int e = flat & 63;
    float* dst = &Wl[f * 66 + e];
    dst[0] = v.x;
    dst[1] = v.y;
    dst[2] = v.z;
    dst[3] = v.w;
  }
  __syncthreads();

  const int wave = tid >> 5;
  const int lane = tid & 31;
  const int m = lane & 15;
  const int half = lane >> 4;
  const int l0 = blockIdx.x * 128 + wave * 16;

  // This lane's A-row pointer: data[l0+m][n][hd][:]
  const float* xrow = data + ((size_t)(l0 + m) * (N * H) + (size_t)n * H + hd) * E;

  // Load A fragments for all 16 K-steps (non-temporal: single-use stream);
  // fold in normalizer; accumulate ||x||^2 on the fly.
  v2f av[16];
  float ss = 0.0f;
#pragma unroll
  for (int kk = 0; kk < 16; ++kk) {
    v2f a = __builtin_nontemporal_load((const v2f*)(xrow + kk * 4 + half * 2));
    a *= NORMALIZER;
    av[kk] = a;
    ss += a[0] * a[0] + a[1] * a[1];
  }
  // Lanes l and l^16 hold complementary halves of row m = l&15.
  const float fullsq = ss + __shfl_xor(ss, 16);

  v8f acc[4] = {};
#pragma unroll
  for (int kk = 0; kk < 16; ++kk) {
    const v2f a = av[kk];
#pragma unroll
    for (int ft = 0; ft < 4; ++ft) {
      const v2f b = *(const v2f*)(&Wl[(ft * 16 + m) * 66 + kk * 4 + half * 2]);
      acc[ft] = __builtin_amdgcn_wmma_f32_16x16x4_f32(
          /*neg_a=*/false, a, /*neg_b=*/false, b,
          /*c_mod=*/(short)0, acc[ft], /*reuse_a=*/false, /*reuse_b=*/false);
    }
  }

  // Epilogue: exp(dot - 0.5*||x||^2) + eps, scattered per C/D layout.
  // Hoist the row-sum shuffle out of the ft loop (8 bpermutes, not 32).
#pragma unroll
  for (int j = 0; j < 8; ++j) {
    const int r = j + half * 8;              // output row within the 16-l tile
    const float s = 0.5f * __shfl(fullsq, r);
    float* orow = out + ((size_t)(l0 + r) * (N * H) + (size_t)n * H + hd) * F + m;
#pragma unroll
    for (int ft = 0; ft < 4; ++ft) {
      const float val = __expf(acc[ft][j] - s) + KEPS;
      __builtin_nontemporal_store(val, orow + ft * 16);
    }
  }
}

extern "C" void kernel_launch(void* const* d_in, const int* in_sizes, int n_in,
                              void* d_out, int out_size, void* d_ws, size_t ws_size,
                              hipStream_t stream) {
  (void)in_sizes; (void)n_in; (void)out_size; (void)ws_size;
  const float* data = (const float*)d_in[0];          // (L, N, H, E) fp32
  const float* proj = (const float*)d_in[1];          // (H, E, E)    fp32
  float* out = (float*)d_out;                         // (L, N, H, F) fp32
  float* W = (float*)d_ws;                            // (H, F, E) scratch, 256 KB

  orth_mgs_kernel<<<dim3(H), dim3(32), 0, stream>>>(proj, W);

  dim3 grid(L / 128, N * H);
  favor_wmma_kernel<<<grid, dim3(256), 0, stream>>>(data, W, out);
}